// ResidualVectorQuantize_84301618086203
// MI455X (gfx1250) — compile-verified
//
#include <hip/hip_runtime.h>
#include <hip/hip_bf16.h>
#include <math.h>

// ---------------- problem constants ----------------
#define RB   16
#define RT   4096
#define RD   512
#define RN   9
#define RK   1024
#define Rd   8
#define BT   (RB * RT)               // 65536 rows

// d_out layout (floats, in return order): z_q, codes, latents, closs, cbloss
#define OFF_ZQ     ((size_t)0)
#define SZ_ZQ      ((size_t)BT * RD)              // 33554432
#define OFF_CODES  (OFF_ZQ + SZ_ZQ)
#define SZ_CODES   ((size_t)BT * RN)              // 589824
#define OFF_LAT    (OFF_CODES + SZ_CODES)
#define SZ_LAT     ((size_t)BT * RN * Rd)         // 4718592
#define OFF_CLOSS  (OFF_LAT + SZ_LAT)
#define OFF_CBLOSS (OFF_CLOSS + 1)

// workspace layout (floats)
#define WS_WINT   ((size_t)0)                      // [9][16][512] padded, zero rows 8..15
#define WS_WOUTT  (WS_WINT + (size_t)RN*16*RD)     // [9][512][8]  (W_out transposed)
#define WS_CBN    (WS_WOUTT + (size_t)RN*RD*Rd)    // [9][1024][8] normalized codebooks
#define WS_LOSS   (WS_CBN + (size_t)RN*RK*Rd)      // 1 float accumulator

typedef __attribute__((ext_vector_type(2))) float    v2f;
typedef __attribute__((ext_vector_type(8))) float    v8f;
typedef __attribute__((ext_vector_type(4))) unsigned v4u;
typedef __attribute__((ext_vector_type(8))) int      v8i;
typedef __attribute__((ext_vector_type(4))) int      v4i;

__device__ __forceinline__ v8f wmma4(v2f a, v2f b, v8f c) {
  // D = A(16x4,f32) * B(4x16,f32) + C(16x16,f32)  -> v_wmma_f32_16x16x4_f32
  return __builtin_amdgcn_wmma_f32_16x16x4_f32(
      /*neg_a=*/false, a, /*neg_b=*/false, b,
      /*c_mod=*/(short)0, c, /*reuse_a=*/false, /*reuse_b=*/false);
}

// 1-D contiguous TDM copy: global -> LDS, nelem f32 elements (nelem < 65536).
// D# per CDNA5 ISA: group0 {count|lds|global|type=2},
// group1 {data_size=4B, tensor_dim0=tile_dim0=nelem, tensor_dim1=1, stride=nelem}.
// This toolchain declares the 6-arg builtin: (v4u, v8i, v4i, v4i, v8i, i32 cpol).
__device__ __forceinline__ void tdm_load_1d(const float* gptr, float* lptr,
                                            unsigned nelem) {
  unsigned long long ga = (unsigned long long)(uintptr_t)gptr;
  unsigned lds = (unsigned)(uintptr_t)lptr;   // generic LDS addr low 32 bits = LDS offset
  v4u g0;
  g0.x = 1u;                                             // count=1 valid descriptor
  g0.y = lds;                                            // lds_addr [63:32]
  g0.z = (unsigned)ga;                                   // global_addr [95:64]
  g0.w = (unsigned)((ga >> 32) & 0x01FFFFFFu) | (2u << 30);  // global_addr hi + type=2
  v8i g1;
  g1[0] = 0x00020000;                                    // data_size = 2 (4 bytes)
  g1[1] = (int)((nelem & 0xFFFFu) << 16);                // tensor_dim0 lo16
  g1[2] = (int)((nelem >> 16) & 0xFFFFu) | (1 << 16);    // tensor_dim0 hi16, tensor_dim1=1
  g1[3] = (int)((nelem & 0xFFFFu) << 16);                // tile_dim0 = nelem
  g1[4] = 1;                                             // tile_dim1 = 1, tile_dim2 unused
  g1[5] = (int)nelem;                                    // tensor_dim0_stride lo32
  g1[6] = 0;
  g1[7] = 0;
  v4i g2 = {0, 0, 0, 0};
  v4i g3 = {0, 0, 0, 0};
  v8i g4 = {0, 0, 0, 0, 0, 0, 0, 0};
  __builtin_amdgcn_tensor_load_to_lds(g0, g1, g2, g3, g4, 0);
}

// ------------- precompute: weight-norm weights + normalized codebooks -------------
__global__ void rvq_prep(const float* __restrict__ in_v, const float* __restrict__ in_g,
                         const float* __restrict__ out_v, const float* __restrict__ out_g,
                         const float* __restrict__ cb, float* __restrict__ ws) {
  const int TASKS_PER_I = 16 + RD + RK;   // 1552
  int tid = blockIdx.x * blockDim.x + threadIdx.x;
  if (tid >= RN * TASKS_PER_I) return;
  int i = tid / TASKS_PER_I;
  int r = tid % TASKS_PER_I;
  float* wsWinT  = ws + WS_WINT;
  float* wsWoutT = ws + WS_WOUTT;
  float* wsCbN   = ws + WS_CBN;

  if (r < 8) {                     // in_proj column n: W_inT[i][n][kk] over kk<512
    int n = r;
    float s = 0.f;
    for (int kk = 0; kk < RD; ++kk) {
      float v = in_v[((size_t)i * RD + kk) * Rd + n];
      s += v * v;
    }
    float sc = in_g[i * Rd + n] / fmaxf(sqrtf(s), 1e-12f);
    for (int kk = 0; kk < RD; ++kk)
      wsWinT[((size_t)i * 16 + n) * RD + kk] = sc * in_v[((size_t)i * RD + kk) * Rd + n];
  } else if (r < 16) {             // zero-pad rows n=8..15
    int n = r;
    for (int kk = 0; kk < RD; ++kk)
      wsWinT[((size_t)i * 16 + n) * RD + kk] = 0.f;
  } else if (r < 16 + RD) {        // out_proj column n: W_outT[i][n][kk] over kk<8
    int n = r - 16;
    float s = 0.f;
    for (int kk = 0; kk < Rd; ++kk) {
      float v = out_v[((size_t)i * Rd + kk) * RD + n];
      s += v * v;
    }
    float sc = out_g[i * RD + n] / fmaxf(sqrtf(s), 1e-12f);
    for (int kk = 0; kk < Rd; ++kk)
      wsWoutT[((size_t)i * RD + n) * Rd + kk] = sc * out_v[((size_t)i * Rd + kk) * RD + n];
  } else {                         // codebook row normalize
    int kidx = r - 16 - RD;
    size_t base = ((size_t)i * RK + kidx) * Rd;
    float s = 0.f;
    for (int c = 0; c < Rd; ++c) { float v = cb[base + c]; s += v * v; }
    float sc = 1.f / fmaxf(sqrtf(s), 1e-12f);
    for (int c = 0; c < Rd; ++c) wsCbN[base + c] = cb[base + c] * sc;
  }
}

// ------------- one residual-VQ stage (launched 9x, sequential on stream) -------------
__global__ __launch_bounds__(256) void rvq_step(
    const float* __restrict__ z, const float* __restrict__ in_b,
    const float* __restrict__ out_b, const float* __restrict__ cb,
    const float* __restrict__ wsWinT, const float* __restrict__ wsWoutT,
    const float* __restrict__ wsCbN, float* __restrict__ wsLoss,
    float* __restrict__ out, int icb) {
  __shared__ float sWoutT[RD * Rd];      // 16 KB
  __shared__ float sCbN[RK * Rd];        // 32 KB
  __shared__ float sEnc[8][16 * Rd];     // 4 KB  (per-wave A-fragment staging)
  __shared__ float sZq[8][16 * Rd];      // 4 KB

  const int tid = threadIdx.x;
  const int wave = tid >> 5, lane = tid & 31;

  // Tensor Data Mover: DMA both LDS tables with two descriptors from wave 0,
  // drain TENSORcnt, then publish block-wide via the barrier.
  if (wave == 0) {
    tdm_load_1d(wsWoutT + (size_t)icb * RD * Rd, sWoutT, RD * Rd);
    tdm_load_1d(wsCbN + (size_t)icb * RK * Rd, sCbN, RK * Rd);
    __builtin_amdgcn_s_wait_tensorcnt(0);
  }
  __syncthreads();

  const int m = lane & 15;               // row within A / col within B
  const bool hi = lane >= 16;
  const int kb = hi ? 2 : 0;             // K sub-offset for A/B fragments
  const int hi8 = hi ? 8 : 0;            // D-matrix row offset for this half
  const size_t row0 = (size_t)blockIdx.x * 128 + wave * 16;

  float* zq = out + OFF_ZQ;

  // ---- stage 1: z_e = (z - z_q) * W_in  via V_WMMA_F32_16X16X4_F32, K=512 ----
  v8f acc = {0.f, 0.f, 0.f, 0.f, 0.f, 0.f, 0.f, 0.f};
  const float* zrow = z  + (row0 + m) * RD;
  const float* qrow = zq + (row0 + m) * RD;
  const float* wrow = wsWinT + ((size_t)icb * 16 + m) * RD;   // W_inT[n][K], n == lane&15
  for (int k = 0; k < RD; k += 4) {
    float2 zz = *(const float2*)(zrow + k + kb);
    float2 qq = *(const float2*)(qrow + k + kb);
    v2f a; a.x = zz.x - qq.x; a.y = zz.y - qq.y;
    v2f b; b.x = wrow[k + kb]; b.y = wrow[k + kb + 1];
    acc = wmma4(a, b, acc);
  }

  // D layout: slot j, this lane -> row (j + hi8), col (lane&15)
  const int n = lane & 15;
  const float bias = (n < Rd) ? in_b[icb * Rd + n] : 0.f;
  float ze[8];
#pragma unroll
  for (int j = 0; j < 8; ++j) ze[j] = acc[j] + bias;

  // latents output
  if (n < Rd) {
#pragma unroll
    for (int j = 0; j < 8; ++j)
      out[OFF_LAT + (row0 + j + hi8) * (RN * Rd) + (size_t)icb * Rd + n] = ze[j];
  }

  // l2 normalize each row (reduce squares across the 16-lane half; cols>=8 are 0)
  float enc[8];
#pragma unroll
  for (int j = 0; j < 8; ++j) {
    float s = ze[j] * ze[j];
    for (int msk = 1; msk < 16; msk <<= 1) s += __shfl_xor(s, msk, 32);
    enc[j] = ze[j] * (1.f / fmaxf(sqrtf(s), 1e-12f));
  }

  if (n < Rd) {
#pragma unroll
    for (int j = 0; j < 8; ++j) sEnc[wave][(j + hi8) * Rd + n] = enc[j];
  }
  __syncthreads();

  // ---- stage 2: cosine scores vs 1024 codes, fused argmax ----
  float best[8]; int bidx[8];
#pragma unroll
  for (int j = 0; j < 8; ++j) { best[j] = -3.4e38f; bidx[j] = 0; }
  const float* encw = sEnc[wave];
  for (int t = 0; t < RK / 16; ++t) {
    v8f sc = {0.f, 0.f, 0.f, 0.f, 0.f, 0.f, 0.f, 0.f};
#pragma unroll
    for (int k = 0; k < Rd; k += 4) {
      v2f a; a.x = encw[m * Rd + k + kb]; a.y = encw[m * Rd + k + kb + 1];
      const float* cbrow = sCbN + (size_t)(t * 16 + n) * Rd;
      v2f b; b.x = cbrow[k + kb]; b.y = cbrow[k + kb + 1];
      sc = wmma4(a, b, sc);
    }
    const int col = t * 16 + n;
#pragma unroll
    for (int j = 0; j < 8; ++j)
      if (sc[j] > best[j]) { best[j] = sc[j]; bidx[j] = col; }
  }
  // reduce winners across the 16-lane half (prefer lower index on ties)
#pragma unroll
  for (int j = 0; j < 8; ++j) {
    for (int msk = 1; msk < 16; msk <<= 1) {
      float ob = __shfl_xor(best[j], msk, 32);
      int   oi = __shfl_xor(bidx[j], msk, 32);
      if (ob > best[j] || (ob == best[j] && oi < bidx[j])) { best[j] = ob; bidx[j] = oi; }
    }
  }

  if (n == 0) {
#pragma unroll
    for (int j = 0; j < 8; ++j)
      out[OFF_CODES + (row0 + j + hi8) * RN + icb] = (float)bidx[j];
  }

  // ---- gather raw codebook vectors + commitment/codebook loss (identical fwd) ----
  float lsum = 0.f;
  if (n < Rd) {
#pragma unroll
    for (int j = 0; j < 8; ++j) {
      float v = cb[((size_t)icb * RK + bidx[j]) * Rd + n];
      sZq[wave][(j + hi8) * Rd + n] = v;
      float dlt = ze[j] - v;
      lsum += dlt * dlt;
    }
  }
  for (int msk = 1; msk < 32; msk <<= 1) lsum += __shfl_xor(lsum, msk, 32);
  if (lane == 0) atomicAdd(wsLoss, lsum);
  __syncthreads();

  // ---- stage 3: proj = z_q_i * W_out, accumulate into z_q (residual is z - z_q) ----
  const float* zqw = sZq[wave];
  for (int t = 0; t < RD / 16; ++t) {
    v8f pr = {0.f, 0.f, 0.f, 0.f, 0.f, 0.f, 0.f, 0.f};
#pragma unroll
    for (int k = 0; k < Rd; k += 4) {
      v2f a; a.x = zqw[m * Rd + k + kb]; a.y = zqw[m * Rd + k + kb + 1];
      const float* worow = sWoutT + (size_t)(t * 16 + n) * Rd;
      v2f b; b.x = worow[k + kb]; b.y = worow[k + kb + 1];
      pr = wmma4(a, b, pr);
    }
    const int col = t * 16 + n;
    const float ob = out_b[icb * RD + col];
#pragma unroll
    for (int j = 0; j < 8; ++j)
      atomicAdd(&zq[(row0 + j + hi8) * RD + col], pr[j] + ob);
  }
}

__global__ void rvq_final(const float* __restrict__ wsLoss, float* __restrict__ out) {
  if (blockIdx.x == 0 && threadIdx.x == 0) {
    float v = wsLoss[0] * (1.f / (float)((size_t)BT * Rd));
    out[OFF_CLOSS]  = v;
    out[OFF_CBLOSS] = v;
  }
}

extern "C" void kernel_launch(void* const* d_in, const int* in_sizes, int n_in,
                              void* d_out, int out_size, void* d_ws, size_t ws_size,
                              hipStream_t stream) {
  const float* z     = (const float*)d_in[0];
  const float* in_v  = (const float*)d_in[1];
  const float* in_g  = (const float*)d_in[2];
  const float* in_b  = (const float*)d_in[3];
  const float* out_v = (const float*)d_in[4];
  const float* out_g = (const float*)d_in[5];
  const float* out_b = (const float*)d_in[6];
  const float* cb    = (const float*)d_in[7];
  float* out = (float*)d_out;
  float* ws  = (float*)d_ws;

  // zero z_q accumulator and loss accumulator (graph-capturable memset nodes)
  (void)hipMemsetAsync(out + OFF_ZQ, 0, SZ_ZQ * sizeof(float), stream);
  (void)hipMemsetAsync(ws + WS_LOSS, 0, sizeof(float), stream);

  const int prep_tasks = RN * (16 + RD + RK);
  rvq_prep<<<(prep_tasks + 255) / 256, 256, 0, stream>>>(in_v, in_g, out_v, out_g, cb, ws);

  for (int i = 0; i < RN; ++i) {
    rvq_step<<<BT / 128, 256, 0, stream>>>(
        z, in_b, out_b, cb,
        ws + WS_WINT, ws + WS_WOUTT, ws + WS_CBN, ws + WS_LOSS,
        out, i);
  }
  rvq_final<<<1, 1, 0, stream>>>(ws + WS_LOSS, out);
}